// GCN_67654324846930
// MI455X (gfx1250) — compile-verified
//
#include <hip/hip_runtime.h>
#include <cstdint>
#include <cstddef>

#define NN   100000   // nodes
#define NE   3200000  // edges
#define FIN  512
#define HID  16
#define NCLS 40

typedef __attribute__((ext_vector_type(2))) float v2f;
typedef __attribute__((ext_vector_type(8))) float v8f;

// ---------------- degree / normalization ----------------

__global__ __launch_bounds__(256) void k_deg_init(float* __restrict__ deg) {
  int n = blockIdx.x * 256 + threadIdx.x;
  if (n < NN) deg[n] = 1.0f;               // self-loop contribution
}

__global__ __launch_bounds__(256) void k_deg_acc(const int* __restrict__ dst,
                                                 float* __restrict__ deg) {
  int e = blockIdx.x * 256 + threadIdx.x;
  if (e < NE) atomicAdd(&deg[dst[e]], 1.0f);
}

__global__ __launch_bounds__(256) void k_rsqrt(float* __restrict__ deg) {
  int n = blockIdx.x * 256 + threadIdx.x;
  if (n < NN) deg[n] = rsqrtf(deg[n]);     // dis = D^{-1/2}
}

// ---------------- layer-1 projection: XW = X @ W1 (WMMA f32) ----------------
// One wave per 16-row tile. K looped in steps of 4 with V_WMMA_F32_16X16X4_F32.
// A (16x4 f32): lanes 0-15 hold K=0(,1), lanes 16-31 hold K=2(,3), M = lane&15.
// B (4x16 f32): VGPR0 lanes 0-15 = row K, lanes 16-31 = row K+2; N = lane&15.
// C/D (16x16 f32): VGPR v -> M = v (+8 for upper lanes), N = lane&15.

__global__ __launch_bounds__(256) void k_gemm1_wmma(const float* __restrict__ X,
                                                    const float* __restrict__ W1,
                                                    float* __restrict__ XW) {
  const int gwave = (blockIdx.x * 256 + (int)threadIdx.x) >> 5;
  const int lane  = threadIdx.x & 31;
  const int row0  = gwave * 16;
  if (row0 >= NN) return;                      // wave-uniform exit (EXEC all-1 inside)

  const int m     = lane & 15;                 // M for A, N for B/C/D
  const int khalf = (lane >> 4) << 1;          // 0 for lanes 0-15, 2 for lanes 16-31
  const float* __restrict__ xrow = X + (size_t)(row0 + m) * FIN;

  v8f c = {};
  for (int k = 0; k < FIN; k += 4) {
    if ((k & 63) == 0) __builtin_prefetch(xrow + k + 128, 0, 0);  // global_prefetch_b8
    v2f a, b;
    a.x = xrow[k + khalf];
    a.y = xrow[k + khalf + 1];
    b.x = W1[(k + khalf)     * HID + m];
    b.y = W1[(k + khalf + 1) * HID + m];
    // D = A*B + C  :  v_wmma_f32_16x16x4_f32
    c = __builtin_amdgcn_wmma_f32_16x16x4_f32(false, a, false, b,
                                              (short)0, c, false, false);
  }

  const int rbase = row0 + ((lane >> 4) << 3); // +8 rows for upper half-wave
#pragma unroll
  for (int v = 0; v < 8; ++v)
    XW[(size_t)(rbase + v) * HID + m] = c[v];
}

// ---------------- layer-1 aggregation: H1 += norm * XW[src] ----------------
// One thread per (edge, feature); edges >= NE are the N self-loops.

__global__ __launch_bounds__(256) void k_agg1(const float* __restrict__ XW,
                                              const int* __restrict__ src,
                                              const int* __restrict__ dst,
                                              const float* __restrict__ dis,
                                              float* __restrict__ H1) {
  size_t t = (size_t)blockIdx.x * 256 + threadIdx.x;
  if (t >= (size_t)(NE + NN) * HID) return;
  int    f = (int)(t & (HID - 1));
  size_t e = t >> 4;
  int s, d;
  if (e < NE) { s = src[e]; d = dst[e]; }
  else        { s = d = (int)(e - NE); }
  float norm = dis[s] * dis[d];
  atomicAdd(&H1[(size_t)d * HID + f], XW[(size_t)s * HID + f] * norm);
}

// ---------------- layer-2 projection fused with bias+ReLU of layer 1 --------

__global__ __launch_bounds__(256) void k_gemm2(const float* __restrict__ H1,
                                               const float* __restrict__ b1,
                                               const float* __restrict__ W2,
                                               float* __restrict__ H2) {
  size_t t = (size_t)blockIdx.x * 256 + threadIdx.x;
  if (t >= (size_t)NN * NCLS) return;
  int    cc = (int)(t % NCLS);
  size_t n  = t / NCLS;
  const float* __restrict__ hrow = H1 + n * HID;
  float acc = 0.0f;
#pragma unroll
  for (int f = 0; f < HID; ++f) {
    float h = fmaxf(hrow[f] + b1[f], 0.0f);    // relu(segsum + b1)
    acc = fmaf(h, W2[f * NCLS + cc], acc);
  }
  H2[n * NCLS + cc] = acc;
}

// ---------------- layer-2 aggregation into d_out ----------------

__global__ __launch_bounds__(256) void k_agg2(const float* __restrict__ H2,
                                              const int* __restrict__ src,
                                              const int* __restrict__ dst,
                                              const float* __restrict__ dis,
                                              float* __restrict__ out) {
  size_t t = (size_t)blockIdx.x * 256 + threadIdx.x;
  if (t >= (size_t)(NE + NN) * NCLS) return;
  int    f = (int)(t % NCLS);
  size_t e = t / NCLS;
  int s, d;
  if (e < NE) { s = src[e]; d = dst[e]; }
  else        { s = d = (int)(e - NE); }
  float norm = dis[s] * dis[d];
  atomicAdd(&out[(size_t)d * NCLS + f], H2[(size_t)s * NCLS + f] * norm);
}

// ---------------- +b2 and log_softmax, in place per row ----------------

__global__ __launch_bounds__(256) void k_lsm(float* __restrict__ out,
                                             const float* __restrict__ b2) {
  int n = blockIdx.x * 256 + threadIdx.x;
  if (n >= NN) return;
  float* row = out + (size_t)n * NCLS;
  float vals[NCLS];
  float mx = -__builtin_inff();
#pragma unroll
  for (int c = 0; c < NCLS; ++c) {
    vals[c] = row[c] + b2[c];
    mx = fmaxf(mx, vals[c]);
  }
  float s = 0.0f;
#pragma unroll
  for (int c = 0; c < NCLS; ++c) s += expf(vals[c] - mx);
  float lse = logf(s) + mx;
#pragma unroll
  for (int c = 0; c < NCLS; ++c) row[c] = vals[c] - lse;
}

// ---------------- launch ----------------

extern "C" void kernel_launch(void* const* d_in, const int* in_sizes, int n_in,
                              void* d_out, int out_size, void* d_ws, size_t ws_size,
                              hipStream_t stream) {
  const float* X  = (const float*)d_in[0];
  const int*   EI = (const int*)  d_in[1];   // [2, NE]: row0 = src, row1 = dst
  const float* W1 = (const float*)d_in[2];
  const float* b1 = (const float*)d_in[3];
  const float* W2 = (const float*)d_in[4];
  const float* b2 = (const float*)d_in[5];
  const int* srcI = EI;
  const int* dstI = EI + NE;
  float* out = (float*)d_out;

  float* ws  = (float*)d_ws;
  float* dis = ws;                          // N
  float* XW  = ws + (size_t)NN;             // 16N
  float* H1  = ws + (size_t)17 * NN;        // 16N (aggregated layer 1)
  float* H2  = ws + (size_t)33 * NN;        // 40N (pre-aggregation layer 2)

  // zero the atomic accumulators every call (deterministic replay)
  hipMemsetAsync(H1,  0, sizeof(float) * (size_t)NN * HID,  stream);
  hipMemsetAsync(out, 0, sizeof(float) * (size_t)NN * NCLS, stream);

  k_deg_init<<<(NN + 255) / 256, 256, 0, stream>>>(dis);
  k_deg_acc <<<(NE + 255) / 256, 256, 0, stream>>>(dstI, dis);
  k_rsqrt   <<<(NN + 255) / 256, 256, 0, stream>>>(dis);

  const int tiles  = (NN + 15) / 16;        // 6250 16-row tiles
  const int blocks = (tiles + 7) / 8;       // 8 waves (256 thr) per block
  k_gemm1_wmma<<<blocks, 256, 0, stream>>>(X, W1, XW);

  size_t t1 = (size_t)(NE + NN) * HID;
  k_agg1<<<(unsigned)((t1 + 255) / 256), 256, 0, stream>>>(XW, srcI, dstI, dis, H1);

  size_t t2 = (size_t)NN * NCLS;
  k_gemm2<<<(unsigned)((t2 + 255) / 256), 256, 0, stream>>>(H1, b1, W2, H2);

  size_t t3 = (size_t)(NE + NN) * NCLS;
  k_agg2<<<(unsigned)((t3 + 255) / 256), 256, 0, stream>>>(H2, srcI, dstI, dis, out);

  k_lsm<<<(NN + 255) / 256, 256, 0, stream>>>(out, b2);
}